// GeneralizedSurfaceLoss_58841051955627
// MI455X (gfx1250) — compile-verified
//
#include <hip/hip_runtime.h>
#include <stdint.h>

// GeneralizedSurfaceLoss, MI455X (gfx1250, wave32).
// probs: [8,4,512,512] f32;  target: [8,512,512] int (harness passes integer as int32).
// Output: single f32 scalar.
//
// Math reduction: chamfer-by-dilation distance == Chebyshev distance to the
// opposite class (capped at 50 -> then 0, which cannot happen for this input).
// Only d in {1,2,3} contributes, so d is computed from a 7x7 neighborhood with
// bit-parallel dilations on 32-pixel words. Trivial (all-empty/all-full) planes
// are reproduced exactly via the mask population count.
//
// CDNA5 paths exercised:
//  - GLOBAL_LOAD_ASYNC_TO_LDS_B128 + S_WAIT_ASYNCCNT: double-buffered DMA of the
//    target rows into LDS, overlapped with ballot-based bitmap construction.
//  - V_WMMA_F32_16X16X4_F32: full-f32 WMMA used as the wave32 lane reducer.
//  - global_prefetch_b8 on the streamed probs.

#define B_      8
#define C_      4
#define H_      512
#define W_      512
#define NCLS    3              // foreground classes 1..3
#define BH      64             // image rows per block
#define ROWS    (BH + 6)       // + 3-row halo each side
#define NWORDS  (W_ / 32)      // 16 words per row
#define NP      (B_ * NCLS)    // 24 planes
#define THREADS 256
#define CHROWS  2              // target rows per async chunk
#define NCHUNK  (ROWS / CHROWS)  // 35
#define EPSF    1e-6f

typedef __attribute__((ext_vector_type(2))) float v2f;
typedef __attribute__((ext_vector_type(8))) float v8f;

// ---------------------------------------------------------------------------
// init: zero the 72 plane accumulators (num[24], den[24], msum[24]) in d_ws.
__global__ void gsl_init(float* __restrict__ acc) {
  int i = threadIdx.x;
  if (i < 3 * NP) acc[i] = 0.0f;
}

// ---------------------------------------------------------------------------
// main: one block per (batch, 64-row band); handles all 3 classes.
__global__ __launch_bounds__(THREADS)
void gsl_main(const float* __restrict__ probs, const int* __restrict__ target,
              float* __restrict__ acc) {
  __shared__ uint32_t maskb[NCLS][ROWS][NWORDS];        // per-class mask bitmaps
  __shared__ uint32_t dbits[NCLS][3][BH][NWORDS];       // d==1,2,3 bit masks
  __shared__ int      tbuf[2][CHROWS * W_];             // async staging (2 x 4KB)

  const int tid  = threadIdx.x;
  const int b    = blockIdx.x / (H_ / BH);
  const int band = blockIdx.x % (H_ / BH);
  const int y0   = band * BH;
  const int lane = tid & 31;

  // ---- Stage 1: DMA target rows into LDS (async, double buffered), then
  // build per-class mask bitmaps with wave32 ballots.
  {
    const int base = (b * H_) * W_;
    auto issue_chunk = [&](int ck, int buf) {
      int i0 = tid * 4;                        // 4 ints = 16 B per lane
      int lr = ck * CHROWS + (i0 >> 9);
      int gy = y0 - 3 + lr;
      gy = gy < 0 ? 0 : (gy > H_ - 1 ? H_ - 1 : gy);   // clamp: address safety only
      int x = i0 & (W_ - 1);
      const int* gsrc = target + base + gy * W_ + x;
      uint32_t lds = (uint32_t)(uintptr_t)&tbuf[buf][i0];
      unsigned long long ga = (unsigned long long)(uintptr_t)gsrc;
      asm volatile("global_load_async_to_lds_b128 %0, %1, off"
                   :: "v"(lds), "v"(ga) : "memory");
    };

    issue_chunk(0, 0);
    for (int ck = 0; ck < NCHUNK; ++ck) {
      if (ck + 1 < NCHUNK) {
        issue_chunk(ck + 1, (ck + 1) & 1);               // prefetch next chunk
        asm volatile("s_wait_asynccnt 0x1" ::: "memory"); // chunk ck is in LDS
      } else {
        asm volatile("s_wait_asynccnt 0x0" ::: "memory"); // drain tail
      }
      __syncthreads();           // all waves' portions of chunk ck visible

      const int* tb = tbuf[ck & 1];
#pragma unroll
      for (int sub = 0; sub < (CHROWS * W_) / THREADS; ++sub) {
        int j  = sub * THREADS + tid;          // 0 .. CHROWS*W_-1
        int lr = ck * CHROWS + (j >> 9);
        int x  = j & (W_ - 1);
        int gy = y0 - 3 + lr;
        int cls = (gy >= 0 && gy < H_) ? (tb[j] - 1) : -1;  // halo -> no pixels
        uint32_t m0 = (uint32_t)__ballot(cls == 0);
        uint32_t m1 = (uint32_t)__ballot(cls == 1);
        uint32_t m2 = (uint32_t)__ballot(cls == 2);
        if ((tid & 31) == 0) {
          int wofs = x >> 5;
          maskb[0][lr][wofs] = m0;
          maskb[1][lr][wofs] = m1;
          maskb[2][lr][wofs] = m2;
        }
      }
      __syncthreads();           // buffer (ck&1) free before it is re-issued
    }
  }

  // ---- Stage 2a: per-word Chebyshev band masks (d exactly 1/2/3).
  // D_k = dilate_k(mask), E_k = dilate_k(complement-inside-image).
  for (int idx = tid; idx < NCLS * BH * NWORDS; idx += THREADS) {
    int cls = idx / (BH * NWORDS);
    int rem = idx - cls * (BH * NWORDS);
    int r   = rem >> 4;
    int w   = rem & (NWORDS - 1);
    int lr  = r + 3;

    uint32_t D1 = 0, D2 = 0, D3 = 0, E1 = 0, E2 = 0, E3 = 0, mctr = 0;
#pragma unroll
    for (int j = -3; j <= 3; ++j) {
      int rr = lr + j;                               // 0..ROWS-1
      uint32_t c  = maskb[cls][rr][w];
      uint32_t l  = (w > 0)          ? maskb[cls][rr][w - 1] : 0u;
      uint32_t rg = (w < NWORDS - 1) ? maskb[cls][rr][w + 1] : 0u;
      int gy = y0 - 3 + rr;
      uint32_t valid = (gy >= 0 && gy < H_) ? 0xffffffffu : 0u;
      uint32_t nc = (~c) & valid;
      uint32_t nl = (w > 0)          ? ((~l)  & valid) : 0u;  // no pixels past edge
      uint32_t nr = (w < NWORDS - 1) ? ((~rg) & valid) : 0u;
      // incremental horizontal dilations (bit i == pixel x = 32*w + i)
      uint32_t h1 = c  | (c >> 1)  | (rg << 31) | (c << 1)  | (l >> 31);
      uint32_t h2 = h1 | (c >> 2)  | (rg << 30) | (c << 2)  | (l >> 30);
      uint32_t h3 = h2 | (c >> 3)  | (rg << 29) | (c << 3)  | (l >> 29);
      uint32_t g1 = nc | (nc >> 1) | (nr << 31) | (nc << 1) | (nl >> 31);
      uint32_t g2 = g1 | (nc >> 2) | (nr << 30) | (nc << 2) | (nl >> 30);
      uint32_t g3 = g2 | (nc >> 3) | (nr << 29) | (nc << 3) | (nl >> 29);
      int aj = (j < 0) ? -j : j;
      if (aj <= 1) { D1 |= h1; E1 |= g1; }
      if (aj <= 2) { D2 |= h2; E2 |= g2; }
      D3 |= h3; E3 |= g3;
      if (j == 0) mctr = c;
    }
    uint32_t m = mctr;
    dbits[cls][0][r][w] = ((~m) & D1)        | (m & E1);
    dbits[cls][1][r][w] = ((~m) & D2 & ~D1)  | (m & E2 & ~E1);
    dbits[cls][2][r][w] = ((~m) & D3 & ~D2)  | (m & E3 & ~E2);
  }
  __syncthreads();

  // ---- Stage 2b: streamed, coalesced pass over probs; per-wave WMMA reduce.
#pragma unroll 1
  for (int cls = 0; cls < NCLS; ++cls) {
    float num = 0.0f, den = 0.0f, ms = 0.0f;
    const float* pp = probs + (((size_t)b * C_ + (cls + 1)) * H_ + y0) * W_;
    for (int idx = tid; idx < BH * W_; idx += THREADS) {
      // idx == r*512 + x (row-major within the band)
      int r = idx >> 9;
      int x = idx & (W_ - 1);
      int wofs = x >> 5;
      uint32_t bit = 1u << (x & 31);
      if (idx + THREADS < BH * W_) __builtin_prefetch(&pp[idx + THREADS], 0, 0);
      uint32_t m = maskb[cls][r + 3][wofs] & bit;
      float mv = m ? 1.0f : 0.0f;
      ms += mv;
      float dd = 0.0f;
      if      (dbits[cls][0][r][wofs] & bit) dd = 1.0f;
      else if (dbits[cls][1][r][wofs] & bit) dd = 2.0f;
      else if (dbits[cls][2][r][wofs] & bit) dd = 3.0f;
      if (dd != 0.0f) {
        float p = pp[idx];
        num += fabsf(p - mv) * dd;
        den += 1.0f;
      }
    }

    // Wave32 reduction via V_WMMA_F32_16X16X4_F32: A = lane partials
    // (A layout: lanes 0-15 -> K0/K1, lanes 16-31 -> K2/K3), B selects
    // column 0 = sum of num, column 1 = sum of den over each lane-half.
    v2f a;    a.x = num;  a.y = den;
    v2f bsel; bsel.x = ((lane & 15) == 0) ? 1.0f : 0.0f;
              bsel.y = ((lane & 15) == 1) ? 1.0f : 0.0f;
    v8f c8 = {};
    c8 = __builtin_amdgcn_wmma_f32_16x16x4_f32(false, a, false, bsel,
                                               (short)0, c8, false, false);
    // Column n of D lives in lane n (rows 0-7) and lane n+16 (rows 8-15).
    float s = c8[0] + c8[1] + c8[2] + c8[3] + c8[4] + c8[5] + c8[6] + c8[7];
    s += __shfl_xor(s, 16, 32);        // lane0: num total, lane1: den total

    v2f a2; a2.x = ms; a2.y = 0.0f;
    v8f c2 = {};
    c2 = __builtin_amdgcn_wmma_f32_16x16x4_f32(false, a2, false, bsel,
                                               (short)0, c2, false, false);
    float s2 = c2[0] + c2[1] + c2[2] + c2[3] + c2[4] + c2[5] + c2[6] + c2[7];
    s2 += __shfl_xor(s2, 16, 32);      // lane0: msum total

    int plane = b * NCLS + cls;
    if (lane == 0) {
      atomicAdd(&acc[plane], s);                 // num
      atomicAdd(&acc[2 * NP + plane], s2);       // msum
    }
    if (lane == 1) atomicAdd(&acc[NP + plane], s);  // den
  }
}

// ---------------------------------------------------------------------------
// final: combine 24 plane losses (single wave32).
__global__ void gsl_final(const float* __restrict__ acc, float* __restrict__ out) {
  int l = threadIdx.x;
  float loss = 0.0f, cnt = 0.0f;
  if (l < NP) {
    float num = acc[l], den = acc[NP + l], ms = acc[2 * NP + l];
    bool trivial = (ms == 0.0f) || (ms == (float)(H_ * W_));  // sdm forced to 0
    if (trivial) {
      cnt = 1.0f;                         // band = whole image (>=32), loss = 0
    } else if (den >= 32.0f) {            // MIN_BAND_VOXELS
      cnt = 1.0f;
      loss = num / (den + EPSF);
    }
  }
  for (int off = 16; off > 0; off >>= 1) {
    loss += __shfl_xor(loss, off, 32);
    cnt  += __shfl_xor(cnt,  off, 32);
  }
  if (l == 0) out[0] = loss / (cnt + EPSF);
}

// ---------------------------------------------------------------------------
extern "C" void kernel_launch(void* const* d_in, const int* in_sizes, int n_in,
                              void* d_out, int out_size, void* d_ws, size_t ws_size,
                              hipStream_t stream) {
  (void)in_sizes; (void)n_in; (void)out_size; (void)ws_size;
  const float* probs  = (const float*)d_in[0];
  const int*   target = (const int*)d_in[1];
  float* acc = (float*)d_ws;            // 72 floats: num[24], den[24], msum[24]

  gsl_init <<<1, 128, 0, stream>>>(acc);
  gsl_main <<<B_ * (H_ / BH), THREADS, 0, stream>>>(probs, target, acc);
  gsl_final<<<1, 32, 0, stream>>>(acc, (float*)d_out);
}